// PreferenceShapley_28037546508448
// MI455X (gfx1250) — compile-verified
//
#include <hip/hip_runtime.h>
#include <math.h>

#define N_ITEMS 200000
#define DIMS    128
#define S_PERMS 128
#define ALPHA_C 0.7f
#define EPS_C   1e-8f

typedef float v2f __attribute__((ext_vector_type(2)));
typedef float v8f __attribute__((ext_vector_type(8)));

// Monotone order-preserving float -> uint encoding for atomicMax.
__device__ __forceinline__ unsigned fenc(float f) {
  unsigned u = __float_as_uint(f);
  return (u & 0x80000000u) ? ~u : (u | 0x80000000u);
}
__device__ __forceinline__ float fdec(unsigned e) {
  unsigned u = (e & 0x80000000u) ? (e & 0x7FFFFFFFu) : ~e;
  return __uint_as_float(u);
}

// ---- kernel 0: tgt norm + reset softmax stats (replay-safe init) ----
__global__ void init_kernel(const float* __restrict__ tgt,
                            float* __restrict__ stats,      // [0]=tnorm [1]=sumexp
                            unsigned* __restrict__ umax) {
  int lane = threadIdx.x;
  float ss = 0.f;
  for (int k = lane; k < DIMS; k += 32) { float v = tgt[k]; ss = fmaf(v, v, ss); }
  #pragma unroll
  for (int o = 16; o; o >>= 1) ss += __shfl_xor(ss, o);
  if (lane == 0) { stats[0] = sqrtf(ss); stats[1] = 0.f; *umax = 0u; }
}

// ---- kernel 1: vals[i] = dot_i * (dot_i/(|row_i||tgt|))^alpha via WMMA ----
// One wave per 16-row tile. A tile staged in LDS (132-float padded rows).
// B = tgt chunk broadcast to all 16 columns -> every D column equals the dots.
__global__ void __launch_bounds__(256)
vals_kernel(const float* __restrict__ neigh, const float* __restrict__ tgt,
            const float* __restrict__ stats, float* __restrict__ vals,
            float* __restrict__ w, int nTiles) {
  __shared__ __align__(16) float lds[8][16 * 132];
  const int lane  = threadIdx.x & 31;
  const int wave  = threadIdx.x >> 5;
  const int hi    = lane >> 4;        // lane group: 0 -> K+0..1, 1 -> K+2..3
  const int row16 = lane & 15;
  float* tile = &lds[wave][0];
  const float tn = fmaxf(stats[0], EPS_C);

  // Preload B (tgt) in WMMA B layout once: VGPR0 = tgt[k0+2*hi], VGPR1 = tgt[k0+1+2*hi]
  v2f bv[32];
  #pragma unroll
  for (int k = 0; k < 32; ++k)
    bv[k] = *(const v2f*)(tgt + 4 * k + 2 * hi);

  const int r = lane >> 1, h = lane & 1;           // load mapping: row, half
  const int wavesTotal = (gridDim.x * blockDim.x) >> 5;
  for (int t = (blockIdx.x * blockDim.x + threadIdx.x) >> 5; t < nTiles;
       t += wavesTotal) {
    // --- coalesced 16x128 tile load -> LDS, sumsq in flight ---
    const float* src = neigh + (size_t)t * (16 * DIMS);
    const float4* srow = (const float4*)(src + r * DIMS + h * 64);
    float* drow = tile + r * 132 + h * 64;
    float ss = 0.f;
    #pragma unroll
    for (int j = 0; j < 16; ++j) {
      float4 v = srow[j];
      ss = fmaf(v.x, v.x, ss); ss = fmaf(v.y, v.y, ss);
      ss = fmaf(v.z, v.z, ss); ss = fmaf(v.w, v.w, ss);
      *(float4*)(drow + 4 * j) = v;
    }
    ss += __shfl_xor(ss, 1);          // lanes 2m,2m+1 hold sumsq(row m)

    // --- 32 x V_WMMA_F32_16X16X4_F32: D[:,n] = dots for all n ---
    v8f acc = {};
    const float* arow = tile + row16 * 132 + 2 * hi;
    #pragma unroll
    for (int k = 0; k < 32; ++k) {
      v2f a = *(const v2f*)(arow + 4 * k);
      acc = __builtin_amdgcn_wmma_f32_16x16x4_f32(false, a, false, bv[k],
                                                  (short)0, acc, false, false);
    }

    // D column 0: lane 0 has rows 0..7 (VGPR j), lane 16 has rows 8..15.
    float vout[8];
    #pragma unroll
    for (int j = 0; j < 8; ++j) {
      float ssj = __shfl(ss, (lane & 16) + 2 * j);   // sumsq(row hi*8+j)
      float dot = acc[j];
      float nn  = fmaxf(sqrtf(ssj), EPS_C);
      float sim = dot / (nn * tn);
      vout[j] = dot * __powf(sim, ALPHA_C);          // sim > 0 (uniform inputs)
    }
    if (row16 == 0) {                                 // lanes 0 and 16 store
      int base = t * 16 + hi * 8;
      float4 lo  = make_float4(vout[0], vout[1], vout[2], vout[3]);
      float4 hi4 = make_float4(vout[4], vout[5], vout[6], vout[7]);
      *(float4*)(vals + base)     = lo;  *(float4*)(vals + base + 4) = hi4;
      *(float4*)(w + base)        = lo;  *(float4*)(w + base + 4)    = hi4;  // w init = vals
    }
  }
}

// ---- kernel 2: w[perm[s,i]] -= vals[perm[s,i-1]]/S  (i>0) ----
__global__ void __launch_bounds__(256)
scatter_kernel(const int* __restrict__ perms, const float* __restrict__ vals,
               float* __restrict__ w) {
  const int s = blockIdx.y;
  const int i = blockIdx.x * blockDim.x + threadIdx.x;
  if (i >= N_ITEMS) return;                 // N%256==64: whole waves drop, no ragged wave
  const int idx = s * N_ITEMS + i;
  const int p  = perms[idx];
  int pp = __shfl_up(p, 1);
  if ((threadIdx.x & 31) == 0 && i > 0) pp = perms[idx - 1];
  if (i > 0) atomicAdd(&w[p], vals[pp] * (-1.0f / S_PERMS));
}

// ---- softmax: max, sum, normalize ----
__global__ void max_kernel(const float* __restrict__ w, unsigned* __restrict__ umax, int n) {
  float m = -INFINITY;
  for (int i = blockIdx.x * blockDim.x + threadIdx.x; i < n; i += gridDim.x * blockDim.x)
    m = fmaxf(m, w[i]);
  #pragma unroll
  for (int o = 16; o; o >>= 1) m = fmaxf(m, __shfl_xor(m, o));
  if ((threadIdx.x & 31) == 0) atomicMax(umax, fenc(m));
}

__global__ void sum_kernel(const float* __restrict__ w, const unsigned* __restrict__ umax,
                           float* __restrict__ sum, int n) {
  const float mx = fdec(*umax);
  float a = 0.f;
  for (int i = blockIdx.x * blockDim.x + threadIdx.x; i < n; i += gridDim.x * blockDim.x)
    a += __expf(w[i] - mx);
  #pragma unroll
  for (int o = 16; o; o >>= 1) a += __shfl_xor(a, o);
  if ((threadIdx.x & 31) == 0) atomicAdd(sum, a);
}

__global__ void final_kernel(float* __restrict__ w, const unsigned* __restrict__ umax,
                             const float* __restrict__ sum, int n) {
  const float mx = fdec(*umax);
  const float rs = 1.0f / *sum;
  int i = blockIdx.x * blockDim.x + threadIdx.x;
  if (i < n) w[i] = __expf(w[i] - mx) * rs;   // in place: w aliases d_out
}

extern "C" void kernel_launch(void* const* d_in, const int* in_sizes, int n_in,
                              void* d_out, int out_size, void* d_ws, size_t ws_size,
                              hipStream_t stream) {
  const float* neigh = (const float*)d_in[0];   // [N, 128]
  const float* tgt   = (const float*)d_in[1];   // [128]
  const int*   perms = (const int*)d_in[2];     // [S, N]
  float* out  = (float*)d_out;                  // doubles as w accumulator
  float* vals = (float*)d_ws;                   // N floats
  float* stats = vals + N_ITEMS;                // [0]=tnorm [1]=sumexp
  unsigned* umax = (unsigned*)(stats + 2);

  init_kernel<<<1, 32, 0, stream>>>(tgt, stats, umax);

  const int nTiles = N_ITEMS / 16;              // 12500, exact
  vals_kernel<<<(nTiles + 7) / 8, 256, 0, stream>>>(neigh, tgt, stats, vals, out, nTiles);

  dim3 sg((N_ITEMS + 255) / 256, S_PERMS);
  scatter_kernel<<<sg, 256, 0, stream>>>(perms, vals, out);

  max_kernel<<<1024, 256, 0, stream>>>(out, umax, N_ITEMS);
  sum_kernel<<<1024, 256, 0, stream>>>(out, umax, stats + 1, N_ITEMS);
  final_kernel<<<(N_ITEMS + 255) / 256, 256, 0, stream>>>(out, umax, stats + 1, N_ITEMS);
}